// Learn_uncertainty_balance_crossEntropy_loss_8452495638824
// MI455X (gfx1250) — compile-verified
//
#include <hip/hip_runtime.h>

typedef __attribute__((ext_vector_type(4))) float v4f;
typedef __attribute__((ext_vector_type(2))) float v2f;
typedef __attribute__((ext_vector_type(8))) float v8f;

#define NTHREADS 256
#define NBLOCKS  2048

// Stage 1: one thread handles one group of 8 rows (64B per input array, coalesced).
// Computes sum over its groups of ce * uncertainty * mask and tree-reduces per block.
__global__ __launch_bounds__(NTHREADS)
void vip_loss_partial_kernel(const float* __restrict__ inp,   // [N,2] student logits
                             const float* __restrict__ tgt,   // [N,2] teacher logits
                             float* __restrict__ partials,    // [NBLOCKS]
                             int numGroups, float invCnt1, float invCnt0) {
    const float INV_LN2 = 1.4426950408889634f;
    int stride = gridDim.x * blockDim.x;
    float acc = 0.0f;

    for (int g = blockIdx.x * blockDim.x + threadIdx.x; g < numGroups; g += stride) {
        size_t base = (size_t)g * 16;
        const v4f* t4 = (const v4f*)(tgt + base);
        const v4f* x4 = (const v4f*)(inp + base);

        // Prefetch next grid-stride iteration (emits global_prefetch_b8).
        int gn = g + stride;
        if (gn < numGroups) {
            __builtin_prefetch(tgt + (size_t)gn * 16, 0, 1);
            __builtin_prefetch(inp + (size_t)gn * 16, 0, 1);
        }

        float tb[16], xb[16];
        #pragma unroll
        for (int q = 0; q < 4; ++q) {
            v4f tv = __builtin_nontemporal_load(t4 + q);   // read-once stream: NT hint
            v4f xv = __builtin_nontemporal_load(x4 + q);
            tb[4*q+0] = tv.x; tb[4*q+1] = tv.y; tb[4*q+2] = tv.z; tb[4*q+3] = tv.w;
            xb[4*q+0] = xv.x; xb[4*q+1] = xv.y; xb[4*q+2] = xv.z; xb[4*q+3] = xv.w;
        }

        // argmax of teacher p1 == argmax of d = t1 - t0 (sigmoid is monotone); first max wins.
        float darr[8];
        #pragma unroll
        for (int j = 0; j < 8; ++j) darr[j] = tb[2*j+1] - tb[2*j];
        int vip = 0; float best = darr[0];
        #pragma unroll
        for (int j = 1; j < 8; ++j) { if (darr[j] > best) { best = darr[j]; vip = j; } }

        #pragma unroll
        for (int j = 0; j < 8; ++j) {
            // Teacher entropy, C=2 closed form: H = |d|*sigmoid(-|d|) + log1p(exp(-|d|))
            float ad = fabsf(darr[j]);
            float e  = __expf(-ad);
            float H  = ad * __fdividef(e, 1.0f + e) + __logf(1.0f + e);
            float unc = 1.0f - H * INV_LN2;   // 1 - H/ln2  ==  1 - sum(p log p)/log(1/C)

            // Student CE vs hard target, C=2: ce = softplus(z) if t==0, softplus(-z) if t==1
            float z  = xb[2*j+1] - xb[2*j];
            float az = fabsf(z);
            float lx = __logf(1.0f + __expf(-az));
            bool isVip = (j == vip);
            float ce = lx + (isVip ? fmaxf(-z, 0.0f) : fmaxf(z, 0.0f));
            float w  = isVip ? invCnt1 : invCnt0;   // exactly one positive per group
            acc += ce * unc * w;
        }
    }

    // Deterministic block tree reduction.
    __shared__ float sdata[NTHREADS];
    int tid = threadIdx.x;
    sdata[tid] = acc;
    __syncthreads();
    #pragma unroll
    for (int s = NTHREADS / 2; s > 0; s >>= 1) {
        if (tid < s) sdata[tid] += sdata[tid + s];
        __syncthreads();
    }
    if (tid == 0) partials[blockIdx.x] = sdata[0];
}

// Stage 2: single block folds NBLOCKS partials; the last 32-lane reduction is done
// with V_WMMA_F32_16X16X4_F32 (A = 16x4 holding the 32 lane values, B = ones).
__global__ __launch_bounds__(NTHREADS)
void vip_loss_final_kernel(const float* __restrict__ partials, int n,
                           float invN, float* __restrict__ out) {
    __shared__ float sdata[NTHREADS];
    int tid = threadIdx.x;
    float s = 0.0f;
    for (int i = tid; i < n; i += NTHREADS) s += partials[i];
    sdata[tid] = s;
    __syncthreads();
    if (tid < 128) sdata[tid] += sdata[tid + 128];
    __syncthreads();
    if (tid < 64)  sdata[tid] += sdata[tid + 64];
    __syncthreads();

    if (tid < 32) {   // wave-uniform: wave 0 only, EXEC all-1s inside
        float val = sdata[tid] + sdata[tid + 32];

        // A (16x4 f32): lanes 0-15 -> A[m,0]=val_m (K=0 in VGPR0), lanes 16-31 -> A[m,2]=val_{m+16}.
        // VGPR1 (K=1 / K=3) = 0. B = all ones => D[m,n] = val_m + val_{m+16} for every n.
        v2f a; a.x = val;  a.y = 0.0f;
        v2f b; b.x = 1.0f; b.y = 1.0f;
        v8f c = {};
        v8f dm = __builtin_amdgcn_wmma_f32_16x16x4_f32(
            /*neg_a=*/false, a, /*neg_b=*/false, b,
            /*c_mod=*/(short)0, c, /*reuse_a=*/false, /*reuse_b=*/false);

        // Column 0: D[0..7,0] live in lane 0 (VGPR 0..7), D[8..15,0] in lane 16.
        float t = dm[0] + dm[1] + dm[2] + dm[3] + dm[4] + dm[5] + dm[6] + dm[7];
        float total = __shfl(t, 0, 32) + __shfl(t, 16, 32);
        if (tid == 0) out[0] = total * invN;
    }
}

extern "C" void kernel_launch(void* const* d_in, const int* in_sizes, int n_in,
                              void* d_out, int out_size, void* d_ws, size_t ws_size,
                              hipStream_t stream) {
    const float* inp = (const float*)d_in[0];   // input_logits  [N,2] f32
    const float* tgt = (const float*)d_in[1];   // target_logits [N,2] f32
    float* out = (float*)d_out;
    float* partials = (float*)d_ws;

    int N = in_sizes[0] / 2;          // flat count = N*C, C=2
    int numGroups = N / 8;            // NI = 8
    // Exactly one positive per group of 8 => cnt1 = N/8, cnt0 = 7N/8 (constants).
    float invCnt1 = 8.0f / (float)N;
    float invCnt0 = 8.0f / (7.0f * (float)N);
    float invN    = 1.0f / (float)N;

    vip_loss_partial_kernel<<<NBLOCKS, NTHREADS, 0, stream>>>(
        inp, tgt, partials, numGroups, invCnt1, invCnt0);
    vip_loss_final_kernel<<<1, NTHREADS, 0, stream>>>(
        partials, NBLOCKS, invN, out);
}